// total_energy_sum_44435731645167
// MI455X (gfx1250) — compile-verified
//
#include <hip/hip_runtime.h>

typedef float v2f __attribute__((ext_vector_type(2)));
typedef float v8f __attribute__((ext_vector_type(8)));

#define N_ATOMS 4096
#define G_GROUPS 32
#define GSZ 128     // atoms per group (N/G)
#define NT 4        // number of types
#define LSTRIDE 34  // padded LDS row stride (floats): even (b64 align) + conflict-free
#define STRIPS 8    // 16-row strips per group

// Cross-kernel partials: one entry per (strip, half) = 256*2. Fully
// overwritten by kernel 1 on every call before kernel 2 reads it.
__device__ float g_partials[G_GROUPS * STRIPS * 2];

// One block = one wave32 = one 16-row strip of one group's 128x128 diagonal
// tile (256 blocks total -> spreads across up to 256 WGPs).
//  - R/F_cut staged to LDS in 16x32 chunks via GLOBAL_LOAD_ASYNC_TO_LDS_B128
//    (coalesced 128B per 8-lane row group, ASYNCcnt-tracked).
//  - Row sums on the matrix pipe with V_WMMA_F32_16X16X4_F32 (full fp32):
//    B-fragment is a ones-column at N=0, so D[:,0] accumulates sum_k A[:,k].
__global__ __launch_bounds__(32) void total_energy_sum_44435731645167_kernel(
    const float* __restrict__ node_attrs,  // N x 4 (one-hot)
    const float* __restrict__ R,           // N x N
    const float* __restrict__ Fcut,        // N x N
    const float* __restrict__ refA,        // 4x4
    const float* __restrict__ refB,        // 4x4
    const float* __restrict__ refC,        // 4x4
    const float* __restrict__ refD,        // 4x4
    const float* __restrict__ refMu,       // 4x4
    const float* __restrict__ atomElecE,   // N
    const float* __restrict__ atomShortE,  // N
    float* __restrict__ out)               // [G energies][N node energies]
{
    __shared__ int   typesLds[GSZ];
    __shared__ float tabLds[5 * NT * NT];      // A,B,C,D,mu packed 16 each
    __shared__ float stage[2 * 16 * LSTRIDE];  // R/F chunk buffers (this wave)

    const int s     = blockIdx.x;      // strip id 0..255
    const int g     = s >> 3;          // group
    const int strip = s & 7;           // strip within group
    const int lane  = threadIdx.x;

    // Stage this group's node types (argmax of one-hot) and the 4x4 tables.
    for (int t = lane; t < GSZ; t += 32) {
        const int node = g * GSZ + t;
        const float a0 = node_attrs[node * 4 + 0];
        const float a1 = node_attrs[node * 4 + 1];
        const float a2 = node_attrs[node * 4 + 2];
        const float a3 = node_attrs[node * 4 + 3];
        int ty = 0; float m = a0;
        if (a1 > m) { m = a1; ty = 1; }
        if (a2 > m) { m = a2; ty = 2; }
        if (a3 > m) { m = a3; ty = 3; }
        typesLds[t] = ty;
    }
    for (int e = lane; e < 80; e += 32) {
        const int which = e >> 4;
        const int idx   = e & 15;
        float v;
        if      (which == 0) v = refA[idx];
        else if (which == 1) v = refB[idx];
        else if (which == 2) v = refC[idx];
        else if (which == 3) v = refD[idx];
        else                 v = refMu[idx];
        tabLds[e] = v;
    }
    __syncthreads();  // single-wave WG: lowers to NOP barrier + counts waits

    // A-fragment layout for 16x4 f32 WMMA:
    //   lanes 0-15 : M = lane,    v0 = K0, v1 = K1
    //   lanes 16-31: M = lane-16, v0 = K2, v1 = K3
    const int rowBase = strip * 16;     // strip's first local row
    const int mLocal  = lane & 15;      // M within strip
    const int half    = lane >> 4;      // K pair selector
    const int iLocal  = rowBase + mLocal;
    const int ti      = typesLds[iLocal];

    // Strip base: global row (g*128 + rowBase), column base g*128.
    const float* Rrow0 = R    + (size_t)(g * GSZ + rowBase) * N_ATOMS + (size_t)g * GSZ;
    const float* Frow0 = Fcut + (size_t)(g * GSZ + rowBase) * N_ATOMS + (size_t)g * GSZ;

    float* Rbuf = stage;
    float* Fbuf = stage + 16 * LSTRIDE;

    // Staging lane map: 8 lanes x 16B cover one 32-col row; 4 rows per op.
    const int rLane = lane >> 3;        // 0..3 : row within quad
    const int cOff  = (lane & 7) * 4;   // 0,4,...,28 : column of this lane's 16B

    // B fragment 4x16: B[k][0] = 1, else 0 -> D[:,0] = row sums.
    const float bsel = (mLocal == 0) ? 1.0f : 0.0f;
    v2f bfrag; bfrag.x = bsel; bfrag.y = bsel;

    v8f acc;
#pragma unroll
    for (int r = 0; r < 8; ++r) acc[r] = 0.0f;

    for (int kc = 0; kc < GSZ; kc += 32) {
        // ---- async-stage this 16x32 chunk of R and F into LDS ----
#pragma unroll
        for (int t = 0; t < 4; ++t) {
            const int rr = t * 4 + rLane;
            const float* gR = Rrow0 + (size_t)rr * N_ATOMS + kc + cOff;
            const float* gF = Frow0 + (size_t)rr * N_ATOMS + kc + cOff;
            const unsigned lR = (unsigned)(uintptr_t)(Rbuf + rr * LSTRIDE + cOff);
            const unsigned lF = (unsigned)(uintptr_t)(Fbuf + rr * LSTRIDE + cOff);
            asm volatile("global_load_async_to_lds_b128 %0, %1, off"
                         :: "v"(lR), "v"(gR) : "memory");
            asm volatile("global_load_async_to_lds_b128 %0, %1, off"
                         :: "v"(lF), "v"(gF) : "memory");
        }
        asm volatile("s_wait_asynccnt 0" ::: "memory");

        // ---- 8 WMMA k-steps over this chunk ----
#pragma unroll
        for (int k4 = 0; k4 < 32; k4 += 4) {
            const int c0 = k4 + (half << 1);               // even column in chunk
            const float2 rv = *(const float2*)(Rbuf + mLocal * LSTRIDE + c0);
            const float2 fv = *(const float2*)(Fbuf + mLocal * LSTRIDE + c0);
            v2f afrag;
#pragma unroll
            for (int u = 0; u < 2; ++u) {
                const int   jLocal = kc + c0 + u;
                const int   te  = ti * NT + typesLds[jLocal];
                const float A   = tabLds[ 0 + te];
                const float B   = tabLds[16 + te];
                const float C   = tabLds[32 + te];
                const float D   = tabLds[48 + te];
                const float mu  = tabLds[64 + te];
                const float r1  = (u == 0) ? rv.x : rv.y;
                const float f   = (u == 0) ? fv.x : fv.y;
                const float ir2 = __builtin_amdgcn_rcpf(r1 * r1);  // v_rcp_f32
                const float ir6 = ir2 * ir2 * ir2;
                const float ir8 = ir6 * ir2;
                float e = (A * __expf(B * (mu - r1)) - C * ir6 - D * ir8) * f;
                e = (iLocal == jLocal) ? 0.0f : e;         // zero diagonal
                afrag[u] = e;
            }
            acc = __builtin_amdgcn_wmma_f32_16x16x4_f32(
                /*neg_a=*/false, afrag, /*neg_b=*/false, bfrag,
                /*c_mod=*/(short)0, acc, /*reuse_a=*/false, /*reuse_b=*/false);
        }
    }

    // Column N=0 of D lives in lanes 0 (M=0..7) and 16 (M=8..15), VGPRs 0..7.
    if (mLocal == 0) {
        const int mBase = half * 8;
        float waveSum = 0.0f;
#pragma unroll
        for (int r = 0; r < 8; ++r) {
            const float rs = acc[r];
            const int   iG = g * GSZ + rowBase + mBase + r;
            out[G_GROUPS + iG] = atomElecE[iG] + 0.5f * rs + atomShortE[iG];
            waveSum += rs;
        }
        g_partials[s * 2 + half] = waveSum;   // = g*16 + strip*2 + half
    }
}

// Kernel 2: per-group reduction of the 16 strip/half partials (deterministic).
__global__ __launch_bounds__(32) void total_energy_sum_finalize_kernel(
    const float* __restrict__ elecE,   // G
    const float* __restrict__ shortE,  // G
    float* __restrict__ out)           // [G energies][N node energies]
{
    const int g = threadIdx.x;  // 0..31
    float sum = 0.0f;
#pragma unroll
    for (int p = 0; p < 16; ++p) sum += g_partials[g * 16 + p];
    out[g] = elecE[g] + 0.5f * sum + shortE[g];
}

extern "C" void kernel_launch(void* const* d_in, const int* in_sizes, int n_in,
                              void* d_out, int out_size, void* d_ws, size_t ws_size,
                              hipStream_t stream) {
    (void)in_sizes; (void)n_in; (void)out_size; (void)d_ws; (void)ws_size;
    const float* node_attrs = (const float*)d_in[0];
    const float* R          = (const float*)d_in[1];
    const float* Fcut       = (const float*)d_in[2];
    const float* refA       = (const float*)d_in[3];
    const float* refB       = (const float*)d_in[4];
    const float* refC       = (const float*)d_in[5];
    const float* refD       = (const float*)d_in[6];
    const float* refMu      = (const float*)d_in[7];
    const float* elecE      = (const float*)d_in[8];
    const float* atomElecE  = (const float*)d_in[9];
    const float* shortE     = (const float*)d_in[10];
    const float* atomShortE = (const float*)d_in[11];
    // d_in[12] = batch: implicit in the block-diagonal structure, unused.
    float* out = (float*)d_out;

    total_energy_sum_44435731645167_kernel<<<dim3(G_GROUPS * STRIPS), dim3(32), 0, stream>>>(
        node_attrs, R, Fcut, refA, refB, refC, refD, refMu,
        atomElecE, atomShortE, out);
    total_energy_sum_finalize_kernel<<<dim3(1), dim3(32), 0, stream>>>(
        elecE, shortE, out);
}